// GCNraw_4088808865982
// MI455X (gfx1250) — compile-verified
//
#include <hip/hip_runtime.h>

typedef float v2f __attribute__((ext_vector_type(2)));
typedef float v8f __attribute__((ext_vector_type(8)));

#define BN_EPS 1e-5f

// ------------------------------------------------------------------
// init: zero the dedup bitmap; deg[i] = 1 (the eye(N) diagonal)
// ------------------------------------------------------------------
__global__ void init_kernel(unsigned* __restrict__ bitmap, size_t bitmapDwords,
                            int* __restrict__ deg, int M) {
  size_t tid = (size_t)blockIdx.x * blockDim.x + threadIdx.x;
  size_t stride = (size_t)gridDim.x * blockDim.x;
  for (size_t i = tid; i < bitmapDwords; i += stride) bitmap[i] = 0u;
  if (tid < (size_t)M) deg[tid] = 1;
}

// ------------------------------------------------------------------
// edges: set bit (i,j); count distinct j != i  (set semantics dedup)
// edge_index is int64 [2, nE]
// ------------------------------------------------------------------
__global__ void edge_kernel(const long long* __restrict__ ei, int nE,
                            unsigned* __restrict__ bitmap, int rowdw,
                            int* __restrict__ deg) {
  int e = blockIdx.x * blockDim.x + threadIdx.x;
  if (e >= nE) return;
  int i = (int)ei[e];
  int j = (int)ei[nE + e];
  unsigned m = 1u << (j & 31);
  unsigned old = atomicOr(&bitmap[(size_t)i * rowdw + (j >> 5)], m);
  if (!(old & m) && i != j) atomicAdd(&deg[i], 1);
}

__global__ void deg_to_float(const int* __restrict__ deg, float* __restrict__ degf, int M) {
  int i = blockIdx.x * blockDim.x + threadIdx.x;
  if (i < M) degf[i] = (float)deg[i];
}

// ------------------------------------------------------------------
// pack W[K,N] -> Wf[K/2][Npad] of v2f: Wf[q][n] = {W[2q][n], W[2q+1][n]},
// zero-padded to Npad (multiple of 64). Removes all guards from the
// GEMM inner loop and makes each B fragment a single b64 load.
// ------------------------------------------------------------------
__global__ void pack_w(const float* __restrict__ W, float* __restrict__ Wf,
                       int K, int N, int Npad) {
  int total = (K / 2) * Npad;
  int stride = gridDim.x * blockDim.x;
  for (int idx = blockIdx.x * blockDim.x + threadIdx.x; idx < total; idx += stride) {
    int q = idx / Npad;
    int n = idx - q * Npad;
    float x0 = 0.0f, x1 = 0.0f;
    if (n < N) {
      x0 = W[(size_t)(2 * q) * N + n];
      x1 = W[(size_t)(2 * q + 1) * N + n];
    }
    Wf[2 * idx] = x0;
    Wf[2 * idx + 1] = x1;
  }
}

// ------------------------------------------------------------------
// GEMM: C[M,N] = relu( (SCALE_A ? deg[m] : 1) * A[M,K] @ W[K,N] + bias )
// One wave owns a 16x64 strip: 4 x (16x16) accumulators via
// V_WMMA_F32_16X16X4_F32, K consumed 4 at a time.
// f32 A 16x4 fragment: lanes 0-15 -> M=lane, K = k+0/k+1 (v0/v1);
//                      lanes 16-31 -> M=lane-16, K = k+2/k+3.
// B fragments come pre-paired from Wf (one v2f load each, no guards).
// C/D: VGPR r holds M=r (lanes 0-15) / M=r+8 (lanes 16-31), N = lane&15.
// ------------------------------------------------------------------
template <bool SCALE_A>
__global__ __launch_bounds__(256) void gemm_wmma_f32(
    const float* __restrict__ A, const v2f* __restrict__ Wf,
    const float* __restrict__ bias, const float* __restrict__ degf,
    float* __restrict__ C, int M, int K, int N, int Npad) {
  const int lane = threadIdx.x & 31;
  const int wid = (int)((blockIdx.x * blockDim.x + threadIdx.x) >> 5);
  const int nWaves = (int)((gridDim.x * blockDim.x) >> 5);
  const int NG = Npad >> 6;                // 64-wide column groups
  const int T = (M >> 4) * NG;             // total wave-tiles
  const int l15 = lane & 15;
  const int qoff = lane >> 4;              // pair index offset: 0 or 1
  const int Kq = K >> 1;                   // number of K-pairs

  for (int tile = wid; tile < T; tile += nWaves) {
    const int mt = tile / NG;
    const int ng = tile - mt * NG;
    const int row = (mt << 4) + l15;
    const v2f* arow = (const v2f*)(A + (size_t)row * K);
    const float sv = SCALE_A ? degf[row] : 1.0f;
    v2f s2;
    s2.x = sv;
    s2.y = sv;
    const int n0 = (ng << 6) + l15;

    v8f acc[4];
#pragma unroll
    for (int t = 0; t < 4; ++t) acc[t] = (v8f){0.f, 0.f, 0.f, 0.f, 0.f, 0.f, 0.f, 0.f};

    const v2f* wbase = Wf + (size_t)qoff * Npad + n0;
    for (int kq = 0; kq < Kq; kq += 2) {
      const v2f a = s2 * arow[kq + qoff];
      const v2f* wrow = wbase + (size_t)kq * Npad;
      acc[0] = __builtin_amdgcn_wmma_f32_16x16x4_f32(false, a, false, wrow[0],
                                                     (short)0, acc[0], false, false);
      acc[1] = __builtin_amdgcn_wmma_f32_16x16x4_f32(false, a, false, wrow[16],
                                                     (short)0, acc[1], false, false);
      acc[2] = __builtin_amdgcn_wmma_f32_16x16x4_f32(false, a, false, wrow[32],
                                                     (short)0, acc[2], false, false);
      acc[3] = __builtin_amdgcn_wmma_f32_16x16x4_f32(false, a, false, wrow[48],
                                                     (short)0, acc[3], false, false);
    }

    const int rbase = (mt << 4) + ((lane >> 4) << 3);
#pragma unroll
    for (int r = 0; r < 8; ++r) {
      float* crow = C + (size_t)(rbase + r) * N;
#pragma unroll
      for (int t = 0; t < 4; ++t) {
        const int n = n0 + (t << 4);
        if (n < N) crow[n] = fmaxf(acc[t][r] + bias[n], 0.0f);
      }
    }
  }
}

// ------------------------------------------------------------------
// BatchNorm (training-mode batch stats, biased variance)
// ------------------------------------------------------------------
__global__ void zero_f32(float* __restrict__ p, int n) {
  int i = blockIdx.x * blockDim.x + threadIdx.x;
  if (i < n) p[i] = 0.0f;
}

// blockDim.x == N; thread c accumulates a row-chunk of column c
__global__ void bn_partial(const float* __restrict__ H, int M, int N, int rowsPerBlock,
                           float* __restrict__ sum, float* __restrict__ sumsq) {
  int c = threadIdx.x;
  int r0 = blockIdx.x * rowsPerBlock;
  int r1 = r0 + rowsPerBlock;
  if (r1 > M) r1 = M;
  float s = 0.0f, q = 0.0f;
  for (int r = r0; r < r1; ++r) {
    float v = H[(size_t)r * N + c];
    s += v;
    q += v * v;
  }
  atomicAdd(&sum[c], s);
  atomicAdd(&sumsq[c], q);
}

// in place: sum -> scale, sumsq -> shift
__global__ void bn_finalize(float* __restrict__ sum, float* __restrict__ sumsq,
                            const float* __restrict__ g, const float* __restrict__ beta,
                            int N, float invM) {
  int c = blockIdx.x * blockDim.x + threadIdx.x;
  if (c >= N) return;
  float mean = sum[c] * invM;
  float var = sumsq[c] * invM - mean * mean;
  float sc = g[c] * rsqrtf(var + BN_EPS);
  sum[c] = sc;
  sumsq[c] = beta[c] - mean * sc;
}

__global__ void bn_apply(float* __restrict__ H, int total, int N,
                         const float* __restrict__ scale, const float* __restrict__ shift) {
  int stride = gridDim.x * blockDim.x;
  for (int i = blockIdx.x * blockDim.x + threadIdx.x; i < total; i += stride) {
    int c = i % N;
    H[i] = H[i] * scale[c] + shift[c];
  }
}

// ------------------------------------------------------------------
// host side
// ------------------------------------------------------------------
template <bool SCALE_A>
static void launch_gemm(const float* A, const float* W, float* Wf, const float* bias,
                        const float* degf, float* C, int M, int K, int N,
                        hipStream_t stream) {
  const int Npad = ((N + 63) / 64) * 64;
  {
    int total = (K / 2) * Npad;
    int blocks = (total + 255) / 256;
    if (blocks > 2048) blocks = 2048;
    pack_w<<<blocks, 256, 0, stream>>>(W, Wf, K, N, Npad);
  }
  const int NG = Npad / 64;
  const int T = (M / 16) * NG;
  const int wavesPerBlock = 8;  // 256 threads
  int blocks = (T + wavesPerBlock - 1) / wavesPerBlock;
  if (blocks < 1) blocks = 1;
  gemm_wmma_f32<SCALE_A><<<blocks, wavesPerBlock * 32, 0, stream>>>(
      A, (const v2f*)Wf, bias, degf, C, M, K, N, Npad);
}

static void run_bn(float* H, int M, int N, const float* g, const float* beta,
                   float* bsum, float* bsq, hipStream_t stream) {
  zero_f32<<<(2 * N + 255) / 256, 256, 0, stream>>>(bsum, 2 * N);  // bsq = bsum + N
  const int BN_BLOCKS = 64;
  int rpb = (M + BN_BLOCKS - 1) / BN_BLOCKS;
  bn_partial<<<BN_BLOCKS, N, 0, stream>>>(H, M, N, rpb, bsum, bsq);
  bn_finalize<<<(N + 255) / 256, 256, 0, stream>>>(bsum, bsq, g, beta, N, 1.0f / (float)M);
  int total = M * N;
  bn_apply<<<2048, 256, 0, stream>>>(H, total, N, bsum, bsq);
}

extern "C" void kernel_launch(void* const* d_in, const int* in_sizes, int n_in,
                              void* d_out, int out_size, void* d_ws, size_t ws_size,
                              hipStream_t stream) {
  const float* x = (const float*)d_in[0];
  const long long* ei = (const long long*)d_in[1];
  const float* W0 = (const float*)d_in[2];
  const float* b0 = (const float*)d_in[3];
  const float* g0 = (const float*)d_in[4];
  const float* be0 = (const float*)d_in[5];
  const float* W1 = (const float*)d_in[6];
  const float* b1 = (const float*)d_in[7];
  const float* g1 = (const float*)d_in[8];
  const float* be1 = (const float*)d_in[9];
  const float* FW0 = (const float*)d_in[10];
  const float* Fb0 = (const float*)d_in[11];
  const float* FW1 = (const float*)d_in[12];
  const float* Fb1 = (const float*)d_in[13];

  const int N1 = in_sizes[3];            // 256
  const int K0 = in_sizes[2] / N1;       // 500
  const int M = in_sizes[0] / K0;        // 10000
  const int nE = in_sizes[1] / 2;        // 320000
  const int N2 = in_sizes[7];            // 256
  const int F0 = in_sizes[11];           // 128
  const int F1 = in_sizes[13];           // 40

  // ---- carve workspace ----
  uintptr_t p = (uintptr_t)d_ws;
  auto carve = [&p](size_t bytes) -> void* {
    uintptr_t r = (p + 255) & ~(uintptr_t)255;
    p = r + bytes;
    return (void*)r;
  };
  const int rowdw = (M + 31) / 32;                         // dwords per bitmap row
  const size_t bitmapDw = (size_t)rowdw * M;               // ~3.13M dwords
  unsigned* bitmap = (unsigned*)carve(bitmapDw * 4);
  int* deg = (int*)carve((size_t)M * 4);
  float* degf = (float*)carve((size_t)M * 4);
  float* h1 = (float*)carve((size_t)M * N1 * 4);
  float* h2 = (float*)carve((size_t)M * N2 * 4);
  float* h3 = (float*)carve((size_t)M * F0 * 4);
  float* bsum = (float*)carve(1024 * 4);                   // [scale|shift] pair region
  float* bsq = bsum + N1;                                  // N1 == N2 == 256
  // packed-weight buffer, reused per layer (max K*Npad = 500*256 floats)
  const size_t wfMax = (size_t)K0 * ((N1 + 63) / 64) * 64; // elements
  float* Wf = (float*)carve(wfMax * 4);
  (void)n_in; (void)ws_size; (void)out_size;

  // ---- degree with set-semantics dedup ----
  init_kernel<<<4096, 256, 0, stream>>>(bitmap, bitmapDw, deg, M);
  edge_kernel<<<(nE + 255) / 256, 256, 0, stream>>>(ei, nE, bitmap, rowdw, deg);
  deg_to_float<<<(M + 255) / 256, 256, 0, stream>>>(deg, degf, M);

  // ---- conv1: h1 = relu((deg*x) @ W0 + b0); BN ----
  launch_gemm<true>(x, W0, Wf, b0, degf, h1, M, K0, N1, stream);
  run_bn(h1, M, N1, g0, be0, bsum, bsq, stream);

  // ---- conv2: h2 = relu((deg*h1) @ W1 + b1); BN ----
  launch_gemm<true>(h1, W1, Wf, b1, degf, h2, M, N1, N2, stream);
  run_bn(h2, M, N2, g1, be1, bsum, bsq, stream);

  // ---- fc0: h3 = relu(h2 @ FW0 + Fb0) ----
  launch_gemm<false>(h2, FW0, Wf, Fb0, nullptr, h3, M, N2, F0, stream);

  // ---- fc1: out = relu(h3 @ FW1 + Fb1) ----
  launch_gemm<false>(h3, FW1, Wf, Fb1, nullptr, (float*)d_out, M, F0, F1, stream);
}